// QuantizedEmbedding_55009941127905
// MI455X (gfx1250) — compile-verified
//
#include <hip/hip_runtime.h>

// QuantizedEmbedding on MI455X (gfx1250):
//   out[b,s,:] = (nibbles(packed[idx]) - zeros[idx,g]) * scales[idx,g]
//
// Pure bandwidth problem (0 matrix FLOPs -> WMMA inapplicable):
//   writes: 32768 tokens * 4KB f32   = 134.2 MB  (streamed, never re-read -> NT stores)
//   reads : 32768 * (512B + 64B+64B) =  ~21 MB   (gather; 65.5MB table fits 192MB L2 -> RT loads)
//   floor @ 23.3 TB/s ~= 6.7 us, store-bound.
//
// Mapping (wave32): 256-thread block, 4 tokens/block.
//   thread t -> 4 consecutive output elems of its token  => one fully
//   contiguous 4KB global_store_b128 nt per token per wave-set,
//   2B packed load per lane (512B/token contiguous),
//   group = t>>4 -> single scale/zero per thread (16-lane broadcast).
//
// v3: branch-free clamped index loads (single s_load_b128 for the 4 uniform
//     indices instead of 4 branchy s_load_b32) -- store keeps the real guard.

typedef __attribute__((ext_vector_type(4))) float v4f;  // native vector: OK for NT store

constexpr int D           = 1024;
constexpr int GROUPS      = 16;    // D / GROUP
constexpr int ROW_BYTES   = 512;   // D / 2 (Q4 packed)
constexpr int THREADS     = 256;
constexpr int TOK_PER_BLK = 4;

__global__ __launch_bounds__(THREADS)
void qembed_dequant_kernel(const int*           __restrict__ indices,
                           const unsigned char* __restrict__ packed,
                           const float*         __restrict__ scales,
                           const float*         __restrict__ zeros,
                           float*               __restrict__ out,
                           int n_tokens)
{
    const int t    = threadIdx.x;
    const int tok0 = blockIdx.x * TOK_PER_BLK;

    // Uniform, branch-free index loads: clamp instead of guard so the
    // compiler can fuse these into one scalar wide load.
    int rows[TOK_PER_BLK];
#pragma unroll
    for (int j = 0; j < TOK_PER_BLK; ++j) {
        int tok = tok0 + j;
        tok = tok < (n_tokens - 1) ? tok : (n_tokens - 1);
        rows[j] = indices[tok];
    }

    // Prefetch later rows' packed data into cache while row 0 is consumed.
    // One lane per 128B line: lanes {0,64,128,192} cover the 512B row.
    if ((t & 63) == 0) {
        const size_t line_off = (size_t)(t >> 6) * 128;
#pragma unroll
        for (int j = 1; j < TOK_PER_BLK; ++j) {
            __builtin_prefetch(packed + (size_t)rows[j] * ROW_BYTES + line_off, 0, 1);
        }
    }

    const int g = t >> 4;   // quant group for this thread's 4 elements

#pragma unroll
    for (int j = 0; j < TOK_PER_BLK; ++j) {
        const int tok = tok0 + j;
        if (tok >= n_tokens) continue;   // only real guard: don't store OOB
        const size_t row = (size_t)rows[j];

        // 2 bytes = 4 nibbles per thread; block reads 512B contiguous.
        const unsigned int pk =
            ((const unsigned short* __restrict__)(packed + row * ROW_BYTES))[t];

        const float s   = scales[row * GROUPS + g];
        const float z   = zeros [row * GROUPS + g];
        const float nzs = -z * s;      // (q - z)*s == fma(q, s, -z*s)

        v4f v;
        v.x = fmaf((float)( pk         & 0xFu), s, nzs);
        v.y = fmaf((float)((pk >>  4u) & 0xFu), s, nzs);
        v.z = fmaf((float)((pk >>  8u) & 0xFu), s, nzs);
        v.w = fmaf((float)((pk >> 12u) & 0xFu), s, nzs);

        // Streaming 134MB output: non-temporal b128 store, fully coalesced
        // (256 threads -> contiguous 4KB), keeps the gather tables hot in L2.
        v4f* o = (v4f*)(out + (size_t)tok * D) + t;
        __builtin_nontemporal_store(v, o);
    }
}

extern "C" void kernel_launch(void* const* d_in, const int* in_sizes, int n_in,
                              void* d_out, int out_size, void* d_ws, size_t ws_size,
                              hipStream_t stream)
{
    const int*           indices = (const int*)          d_in[0];  // [B*S] int32
    const unsigned char* packed  = (const unsigned char*)d_in[1];  // [VOCAB, D/2] u8
    const float*         scales  = (const float*)        d_in[2];  // [VOCAB, D/GROUP] f32
    const float*         zeros   = (const float*)        d_in[3];  // [VOCAB, D/GROUP] f32
    float*               out     = (float*)              d_out;    // [B*S, D] f32

    const int n_tokens = in_sizes[0];
    const int grid = (n_tokens + TOK_PER_BLK - 1) / TOK_PER_BLK;

    qembed_dequant_kernel<<<grid, THREADS, 0, stream>>>(
        indices, packed, scales, zeros, out, n_tokens);
}